// PyTorchGMM_22239340658815
// MI455X (gfx1250) — compile-verified
//
#include <hip/hip_runtime.h>
#include <math.h>

// ext-vector types matching the CDNA5 WMMA builtins
typedef __attribute__((ext_vector_type(2))) float v2f;
typedef __attribute__((ext_vector_type(8))) float v8f;

#define GMM_D   32
#define ICF_C   (GMM_D + GMM_D*(GMM_D-1)/2)   // 528
#define APK     (18*64)                        // floats per k in packed-A workspace
#define KSTAGE  8                              // components staged in LDS at once

// ---------------------------------------------------------------------------
// Prep: build M_k = diag(exp(icf[:d])) + L_k (strictly lower, col-major tail),
// v_k = M_k @ mu_k, pack (M_k | -v_k) into the f32 16x16x4 WMMA A-operand
// lane layout: chunk c = s*2+t holds A rows i=16t+(lane&15), cols 4s+2*(lane>>4)+{0,1}.
// ---------------------------------------------------------------------------
__global__ void gmm_prep(const float* __restrict__ alphas,
                         const float* __restrict__ means,
                         const float* __restrict__ icf,
                         float* __restrict__ apack,
                         float* __restrict__ ab) {
  const int k    = blockIdx.x;
  const int lane = threadIdx.x;               // 32 lanes, one wave
  const float* ick = icf + (size_t)k * ICF_C;

  __shared__ float M[GMM_D][GMM_D + 1];
  __shared__ float vneg[GMM_D];

  const int i = lane;                          // this lane owns row i
  const float qd = __expf(ick[i]);
  float dot = 0.f;
  for (int j = 0; j < GMM_D; ++j) {
    float v;
    if (j == i) {
      v = qd;
    } else if (j < i) {
      const int off = j*(GMM_D-1) - (j*(j-1))/2;       // col-major tril offset
      v = ick[GMM_D + off + (i - j - 1)];
    } else {
      v = 0.f;
    }
    M[i][j] = v;
    dot += v * means[k*GMM_D + j];
  }
  vneg[i] = -dot;
  __syncthreads();

  float* outk = apack + (size_t)k * APK;
  const int half = lane >> 4, r = lane & 15;
  #pragma unroll
  for (int s = 0; s < 8; ++s) {
    #pragma unroll
    for (int t = 0; t < 2; ++t) {
      const int i2 = 16*t + r;
      const int j0 = 4*s + 2*half;
      outk[(s*2+t)*64 + lane*2 + 0] = M[i2][j0];
      outk[(s*2+t)*64 + lane*2 + 1] = M[i2][j0+1];
    }
  }
  #pragma unroll
  for (int t = 0; t < 2; ++t) {               // extra K-slice: A col 32 = -v_k
    const int i2 = 16*t + r;
    outk[(16+t)*64 + lane*2 + 0] = (half == 0) ? vneg[i2] : 0.f;
    outk[(16+t)*64 + lane*2 + 1] = 0.f;
  }
  if (lane == 0) {
    float s = 0.f;
    for (int j = 0; j < GMM_D; ++j) s += ick[j];
    ab[k] = alphas[k] + s;                     // alpha_k + sum_qs_k
  }
}

__global__ void gmm_zero(float* p) { p[0] = 0.f; }

// ---------------------------------------------------------------------------
// Main: each wave owns a 16-point tile. B operand = x^T tile (lane = point
// column l&15, lane-half = K-pair), constant across k. Per k: 18 WMMAs,
// lane-local square-sum + one xor-16 shuffle, streaming logsumexp.
// ---------------------------------------------------------------------------
__global__ void __launch_bounds__(256)
gmm_main(const float* __restrict__ x,
         const float* __restrict__ apack,
         const float* __restrict__ ab,
         float* __restrict__ slse,
         int N, int K) {
  __shared__ alignas(16) float ldsA[KSTAGE * APK];   // 36 KB
  __shared__ float ldsAB[64];
  __shared__ float ldsRed[8];

  const int tid  = threadIdx.x;
  const int wave = tid >> 5;
  const int lane = tid & 31;
  const int half = lane >> 4;
  const int r    = lane & 15;

  const int tile  = blockIdx.x * 8 + wave;
  const int gp    = tile * 16 + r;              // point this lane's column holds
  const bool valid = gp < N;

  if (tid < K) ldsAB[tid] = ab[tid];

  // B operand registers: bx[s] = { x[gp][4s+2*half], x[gp][4s+2*half+1] }
  v2f bx[8];
  const float* xr = x + (size_t)gp * GMM_D;
  #pragma unroll
  for (int s = 0; s < 8; ++s) {
    if (valid) {
      bx[s] = *(const v2f*)(xr + 4*s + 2*half);
    } else {
      bx[s].x = 0.f; bx[s].y = 0.f;
    }
  }
  v2f bones; bones.x = (half == 0) ? 1.f : 0.f; bones.y = 0.f; // ones K-row

  float mx = -INFINITY, ssum = 0.f;

  for (int kbase = 0; kbase < K; kbase += KSTAGE) {
    __syncthreads();
    { // cooperative stage of KSTAGE packed matrices into LDS
      const float4* src = (const float4*)(apack + (size_t)kbase * APK);
      float4* dst = (float4*)ldsA;
      for (int i = tid; i < (KSTAGE*APK)/4; i += 256) dst[i] = src[i];
    }
    __syncthreads();

    for (int kk = 0; kk < KSTAGE; ++kk) {
      const v2f* Ak = (const v2f*)(ldsA + kk * APK);
      v8f acc0 = {0,0,0,0,0,0,0,0};
      v8f acc1 = {0,0,0,0,0,0,0,0};
      #pragma unroll
      for (int s = 0; s < 8; ++s) {
        v2f a0 = Ak[(s*2+0)*32 + lane];
        v2f a1 = Ak[(s*2+1)*32 + lane];
        acc0 = __builtin_amdgcn_wmma_f32_16x16x4_f32(false, a0, false, bx[s],
                                                     (short)0, acc0, false, false);
        acc1 = __builtin_amdgcn_wmma_f32_16x16x4_f32(false, a1, false, bx[s],
                                                     (short)0, acc1, false, false);
      }
      { // -v_k shift folded in as extra K-step
        v2f a0 = Ak[16*32 + lane];
        v2f a1 = Ak[17*32 + lane];
        acc0 = __builtin_amdgcn_wmma_f32_16x16x4_f32(false, a0, false, bones,
                                                     (short)0, acc0, false, false);
        acc1 = __builtin_amdgcn_wmma_f32_16x16x4_f32(false, a1, false, bones,
                                                     (short)0, acc1, false, false);
      }
      // ||Lx||^2 for this lane's point: 16 dims here + 16 dims in partner lane
      float sq = 0.f;
      #pragma unroll
      for (int v = 0; v < 8; ++v) sq += acc0[v]*acc0[v] + acc1[v]*acc1[v];
      sq += __shfl_xor(sq, 16);

      const float inner = ldsAB[kbase + kk] - 0.5f * sq;
      const float m2 = fmaxf(mx, inner);
      ssum = ssum * __expf(mx - m2) + __expf(inner - m2);
      mx = m2;
    }
  }

  // per-point logsumexp -> block sum -> global atomic
  float contrib = (half == 0 && valid) ? (mx + __logf(ssum)) : 0.f;
  contrib += __shfl_xor(contrib, 1);
  contrib += __shfl_xor(contrib, 2);
  contrib += __shfl_xor(contrib, 4);
  contrib += __shfl_xor(contrib, 8);
  if (lane == 0) ldsRed[wave] = contrib;
  __syncthreads();
  if (tid == 0) {
    float s = 0.f;
    #pragma unroll
    for (int w = 0; w < 8; ++w) s += ldsRed[w];
    atomicAdd(slse, s);
  }
}

// ---------------------------------------------------------------------------
// Final: Wishart prior + constants + alpha logsumexp + multigammaln.
// ---------------------------------------------------------------------------
__global__ void gmm_final(const float* __restrict__ alphas,
                          const float* __restrict__ icf,
                          const float* __restrict__ wg,
                          const float* __restrict__ wm,
                          const float* __restrict__ slse,
                          float* __restrict__ out,
                          int N, int K) {
  __shared__ float shw[64];
  const int t = threadIdx.x;                  // K threads
  const float* ick = icf + (size_t)t * ICF_C;
  float sumq = 0.f, q2 = 0.f, l2 = 0.f;
  for (int i = 0; i < GMM_D; ++i) {
    const float v = ick[i];
    sumq += v;
    const float e = __expf(v);
    q2 += e*e;
  }
  for (int i = GMM_D; i < ICF_C; ++i) { const float v = ick[i]; l2 += v*v; }
  const float g = wg[0], m = wm[0];
  shw[t] = 0.5f*g*g*(q2 + l2) - m*sumq;
  __syncthreads();
  if (t == 0) {
    float outw = 0.f, amax = -INFINITY;
    for (int k2 = 0; k2 < K; ++k2) { outw += shw[k2]; amax = fmaxf(amax, alphas[k2]); }
    float se = 0.f;
    for (int k2 = 0; k2 < K; ++k2) se += __expf(alphas[k2] - amax);
    const float lse_a = amax + __logf(se);

    const float dF = (float)GMM_D;
    const float nw = dF + m + 1.f;
    const float C  = nw * dF * __logf(g / sqrtf(2.f));
    const float a  = 0.5f * nw;
    float mg = dF*(dF-1.f)*0.25f * __logf((float)M_PI);
    for (int j = 1; j <= GMM_D; ++j) mg += lgammaf(a + 0.5f*(1.f - (float)j));
    const float CONSTANT = -(float)N * dF * 0.5f * __logf(2.f*(float)M_PI);
    const float prior = outw - (float)K * (C - mg);
    out[0] = CONSTANT + slse[0] - (float)N * lse_a + prior;
  }
}

// ---------------------------------------------------------------------------
extern "C" void kernel_launch(void* const* d_in, const int* in_sizes, int n_in,
                              void* d_out, int out_size, void* d_ws, size_t ws_size,
                              hipStream_t stream) {
  const float* alphas = (const float*)d_in[0];
  const float* means  = (const float*)d_in[1];
  const float* icf    = (const float*)d_in[2];
  const float* x      = (const float*)d_in[3];
  const float* wgam   = (const float*)d_in[4];
  const float* wm     = (const float*)d_in[5];

  const int K = in_sizes[0];                 // 64
  const int N = in_sizes[3] / GMM_D;         // 50000

  float* apack = (float*)d_ws;               // K * APK floats (~295 KB)
  float* ab    = apack + (size_t)K * APK;    // K floats
  float* slse  = ab + K;                     // 1 float accumulator

  gmm_prep<<<K, 32, 0, stream>>>(alphas, means, icf, apack, ab);
  gmm_zero<<<1, 1, 0, stream>>>(slse);

  const int nTiles = (N + 15) / 16;
  const int blocks = (nTiles + 7) / 8;       // 8 waves/block, 1 tile/wave
  gmm_main<<<blocks, 256, 0, stream>>>(x, apack, ab, slse, N, K);

  gmm_final<<<1, K, 0, stream>>>(alphas, icf, wgam, wm, slse, (float*)d_out, N, K);
}